// ProtoNet_53206054863437
// MI455X (gfx1250) — compile-verified
//
#include <hip/hip_runtime.h>

// Problem constants (match reference setup_inputs)
#define KK 64
#define DD 320
#define NN 2048
#define MM 128
#define SS (KK * MM) /* 8192 */
#define EPSN 1e-12f
#define JT 8 /* feature columns per scatter block */

typedef float v2f __attribute__((ext_vector_type(2)));
typedef float v8f __attribute__((ext_vector_type(8)));

// ---------------------------------------------------------------- zero init
__global__ void k_zero(float* counts, float* lblsum, float* cls, float* u, float* scal) {
    int i = blockIdx.x * blockDim.x + threadIdx.x;
    if (i < SS) { counts[i] = 0.f; lblsum[i] = 0.f; cls[i] = 0.f; }
    if (i < DD) u[i] = 0.f;
    if (i < 8)  scal[i] = 0.f;
}

// -------------------------------------------- per-cloud count/label histogram
__global__ void k_count(const float* __restrict__ label, const int* __restrict__ clab,
                        float* counts, float* lblsum) {
    __shared__ float cnt[MM];
    __shared__ float lb[MM];
    int k = blockIdx.x, tid = threadIdx.x;
    if (tid < MM) { cnt[tid] = 0.f; lb[tid] = 0.f; }
    __syncthreads();
    for (int n = tid; n < NN; n += blockDim.x) {
        int seg = clab[k * NN + n];
        atomicAdd(&cnt[seg], 1.0f);               // ds_add_f32
        atomicAdd(&lb[seg], label[k * NN + n]);
    }
    __syncthreads();
    if (tid < MM) { counts[k * MM + tid] = cnt[tid]; lblsum[k * MM + tid] = lb[tid]; }
}

// -------------------------------------- feature scatter-sum (no global atomics)
// block = (k, jt): owns feature columns [jt*JT, jt*JT+JT) of cloud k.
__global__ void k_scatter(const float* __restrict__ feat, const int* __restrict__ clab,
                          float* __restrict__ macc /* d_out means region, gets sums */) {
    __shared__ float acc[JT * MM]; // 4 KB
    int k = blockIdx.x, jt = blockIdx.y, tid = threadIdx.x;
    for (int i = tid; i < JT * MM; i += blockDim.x) acc[i] = 0.f;
    __syncthreads();
    int j0 = jt * JT;
    for (int n = tid; n < NN; n += blockDim.x) {
        int seg = clab[k * NN + n];
#pragma unroll
        for (int j = 0; j < JT; ++j) {
            float v = feat[(size_t)(k * DD + j0 + j) * NN + n]; // coalesced over n
            atomicAdd(&acc[j * MM + seg], v);                   // ds_add_f32
        }
    }
    __syncthreads();
    for (int i = tid; i < JT * MM; i += blockDim.x) {
        int j = i / MM, m = i % MM;
        macc[(size_t)(k * MM + m) * DD + j0 + j] = acc[i]; // exclusive owner -> plain store
    }
}

// ---------------------- finalize: means (in place), L2 norm, fn, masks, maxima
__global__ void __launch_bounds__(DD) k_final(const float* __restrict__ counts,
                                              const float* __restrict__ lblsum,
                                              float* __restrict__ means, // d_out, holds sums
                                              float* __restrict__ fnb,
                                              float* __restrict__ cls,
                                              float* scal) {
    __shared__ float red[512];
    int s = blockIdx.x, j = threadIdx.x; // 320 threads
    float c = counts[s];
    float denom = fmaxf(c, 1.0f);
    float mean = means[(size_t)s * DD + j] / denom;
    means[(size_t)s * DD + j] = mean;
    red[j] = mean * mean;
    if (j < 512 - DD) red[j + DD] = 0.f;
    __syncthreads();
    for (int off = 256; off > 0; off >>= 1) {
        if (j < off) red[j] += red[j + off];
        __syncthreads();
    }
    float nrm = sqrtf(red[0]);
    float lblmean = lblsum[s] / denom;
    float cm = (lblmean > 0.5f && c > 0.f) ? 1.0f : 0.0f;
    fnb[(size_t)s * DD + j] = cm * (mean / fmaxf(nrm, EPSN));
    if (j == 0) {
        cls[s] = cm;
        float sz = c * cm; // sizes >= 0 -> uint-bit compare == float compare
        atomicMax((unsigned int*)&scal[0], __float_as_uint(sz));
        atomicAdd(&scal[1], cm); // n_valid accumulation
    }
}

// ------------------------------------- u = fn^T @ weight  (factored gram trick)
__global__ void __launch_bounds__(DD) k_uvec(const float* __restrict__ fnb,
                                             const float* __restrict__ counts,
                                             const float* __restrict__ cls,
                                             const float* __restrict__ scal,
                                             float* u) {
    int j = threadIdx.x;       // 320 feature lanes, coalesced
    int blk = blockIdx.x;      // 32 blocks x 256 segments
    float maxsz = fmaxf(__uint_as_float(((const unsigned int*)scal)[0]), 1.0f);
    float partial = 0.f;
    int sbase = blk * (SS / 32);
    for (int i = 0; i < SS / 32; ++i) {
        int s = sbase + i;
        float w = counts[s] * cls[s] / maxsz;
        partial = fmaf(fnb[(size_t)s * DD + j], w, partial);
    }
    atomicAdd(&u[j], partial);
}

// --------------------- sim_score[s] = fn[s,:] . u   via V_WMMA_F32_16X16X4_F32
// One wave per 16 rows. A = 16x4 fn tile (ISA layout: lanes 0-15 K={0,1},
// lanes 16-31 K={2,3}, M = lane&15). B = 4x16 with u[k] replicated per column.
__global__ void __launch_bounds__(32) k_sim(const float* __restrict__ fnb,
                                            const float* __restrict__ u,
                                            const float* __restrict__ cls,
                                            float* __restrict__ sim, float* scal) {
    int lane = threadIdx.x;
    int s0 = blockIdx.x * 16;
    int kof = (lane >> 4) * 2;
    const float* arow = fnb + (size_t)(s0 + (lane & 15)) * DD;
    v8f c = {};
    for (int kk = 0; kk < DD; kk += 4) {
        v2f a, b;
        a.x = arow[kk + kof];
        a.y = arow[kk + kof + 1];
        b.x = u[kk + kof];
        b.y = u[kk + kof + 1];
        c = __builtin_amdgcn_wmma_f32_16x16x4_f32(false, a, false, b, (short)0, c,
                                                  false, false);
    }
    // D layout: VGPR r -> (lane 0-15: M=r, lane 16-31: M=8+r); take column N=0.
    if (lane == 0 || lane == 16) {
        int base = s0 + ((lane >> 4) << 3);
        float local = 0.f;
#pragma unroll
        for (int r = 0; r < 8; ++r) {
            float v = c[r];
            sim[base + r] = v;
            local += v * cls[base + r];
        }
        atomicAdd(&scal[2], local); // sum(sim * cls)
    }
}

// ------------------------------------------------------------------ clean mask
__global__ void k_clean(const float* __restrict__ sim, const float* __restrict__ cls,
                        const float* __restrict__ scal, float* __restrict__ clean) {
    int s = blockIdx.x * blockDim.x + threadIdx.x;
    if (s >= SS) return;
    float mean_score = scal[2] / fmaxf(scal[1], 1.0f);
    clean[s] = (sim[s] > mean_score && cls[s] > 0.f) ? 1.0f : 0.0f;
}

// ============================================================================
extern "C" void kernel_launch(void* const* d_in, const int* in_sizes, int n_in,
                              void* d_out, int out_size, void* d_ws, size_t ws_size,
                              hipStream_t stream) {
    const float* feat  = (const float*)d_in[0]; // (K, d, N)
    const float* label = (const float*)d_in[1]; // (K, N)
    const int*   clab  = (const int*)d_in[2];   // (K, N)

    float* out   = (float*)d_out;
    float* means = out;                 // (S, d) — also used as sum accumulator
    float* sim   = out + (size_t)SS * DD;
    float* clean = sim + SS;

    float* counts = (float*)d_ws;       // S
    float* lblsum = counts + SS;        // S
    float* cls    = lblsum + SS;        // S
    float* fnb    = cls + SS;           // S*d
    float* u      = fnb + (size_t)SS * DD; // d
    float* scal   = u + DD;             // [0]=max size bits, [1]=n_valid, [2]=score sum

    k_zero<<<(SS + 255) / 256, 256, 0, stream>>>(counts, lblsum, cls, u, scal);
    k_count<<<KK, 256, 0, stream>>>(label, clab, counts, lblsum);
    k_scatter<<<dim3(KK, DD / JT), 256, 0, stream>>>(feat, clab, means);
    k_final<<<SS, DD, 0, stream>>>(counts, lblsum, means, fnb, cls, scal);
    k_uvec<<<32, DD, 0, stream>>>(fnb, counts, cls, scal, u);
    k_sim<<<SS / 16, 32, 0, stream>>>(fnb, u, cls, sim, scal);
    k_clean<<<(SS + 255) / 256, 256, 0, stream>>>(sim, cls, scal, clean);
}